// LayoutNet_44899588112504
// MI455X (gfx1250) — compile-verified
//
#include <hip/hip_runtime.h>

// ---------------------------------------------------------------------------
// GraphSAGE forward on MI455X (gfx1250), fp32 via V_WMMA_F32_16X16X4_F32.
// Wave32. LDS-tiled GEMMs: block = 128 threads = 4 waves = 64 output rows.
// Inner loop is branch-free: ds_load (A,B fragments) + v_wmma only.
// Staging uses GLOBAL_LOAD_ASYNC_TO_LDS_B128 when available.
// ---------------------------------------------------------------------------

typedef __attribute__((ext_vector_type(2))) float v2f;
typedef __attribute__((ext_vector_type(4))) float v4f;
typedef __attribute__((ext_vector_type(8))) float v8f;
typedef __attribute__((ext_vector_type(4))) int v4i;

#define HDIM 64
#define NGRAPH 64
#define KEMB 192          // 140 + 32 + 18 padded to multiple of KC granules
#define KREAL 190
#define LOUT 128
#define KC 64             // K-chunk staged in LDS per iteration
#define GEMM_ROWS 64      // output rows per block (4 waves x 16)
#define GEMM_THREADS 128
#define APAD (KC + 4)     // padded LDS row stride (floats) -> conflict-free

#if defined(__has_builtin)
#if __has_builtin(__builtin_amdgcn_global_load_async_to_lds_b128) && \
    __has_builtin(__builtin_amdgcn_s_wait_asynccnt)
#define USE_ASYNC_LDS 1
#endif
#endif

__device__ __forceinline__ void cp16_g2l(float* ldst, const float* gsrc) {
#if USE_ASYNC_LDS
  __builtin_amdgcn_global_load_async_to_lds_b128(
      (__attribute__((address_space(1))) v4i*)(gsrc),
      (__attribute__((address_space(3))) v4i*)(ldst), 0, 0);
#else
  *(v4f*)ldst = *(const v4f*)gsrc;
#endif
}

__device__ __forceinline__ void stage_fence() {
#if USE_ASYNC_LDS
  __builtin_amdgcn_s_wait_asynccnt(0);
#endif
  __syncthreads();
}

__device__ __forceinline__ v8f wmma_k4(v2f a, v2f b, v8f c) {
  // D = A(16x4,f32) * B(4x16,f32) + C(16x16,f32)
  return __builtin_amdgcn_wmma_f32_16x16x4_f32(
      false, a, false, b, (short)0, c, false, false);
}

// ------------------------------ utility fills ------------------------------

__global__ void fill_f32_kernel(float* __restrict__ p, float v, long long n) {
  long long i = (long long)blockIdx.x * blockDim.x + threadIdx.x;
  long long stride = (long long)gridDim.x * blockDim.x;
  for (; i < n; i += stride) p[i] = v;
}

__global__ void pool_init_kernel(float* __restrict__ sums, int* __restrict__ maxi,
                                 float* __restrict__ cnt) {
  int i = blockIdx.x * blockDim.x + threadIdx.x;
  if (i < NGRAPH * HDIM) {
    sums[i] = 0.0f;
    maxi[i] = (int)0x80000000;  // encoded -inf
  }
  if (i < NGRAPH) cnt[i] = 0.0f;
}

// ----------------------- input prep (branch-free GEMMs) --------------------
// XC[node, 0:192] = concat(node_feat, op_emb[opcode], cfg, 0, 0)

__global__ void concat_pad_kernel(const float* __restrict__ nf,
                                  const float* __restrict__ cf,
                                  const int* __restrict__ opc,
                                  const float* __restrict__ opE,
                                  float* __restrict__ xc) {
  int node = blockIdx.x;
  int k = threadIdx.x;  // 0..191
  float v;
  if (k < 140)      v = nf[(long long)node * 140 + k];
  else if (k < 172) v = opE[(long long)opc[node] * 32 + (k - 140)];
  else if (k < 190) v = cf[(long long)node * 18 + (k - 172)];
  else              v = 0.0f;
  xc[(long long)node * KEMB + k] = v;
}

// W2 = lin_w zero-padded from 190x128 to 192x128 (same row-major layout)
__global__ void pad_w_kernel(const float* __restrict__ w, float* __restrict__ w2) {
  int i = blockIdx.x * blockDim.x + threadIdx.x;
  if (i < KEMB * LOUT) w2[i] = (i < KREAL * LOUT) ? w[i] : 0.0f;
}

// ------------------------------ degree -------------------------------------

__global__ void degree_kernel(const int* __restrict__ dst, float* __restrict__ deg,
                              int n_edges) {
  int e = blockIdx.x * blockDim.x + threadIdx.x;
  if (e < n_edges) atomicAdd(&deg[dst[e]], 1.0f);
}

// --------------------- LDS-tiled GEMM + ReLU -------------------------------
// out = relu(A(NxK) @ W(K x NT*16) + bias),  K % 64 == 0

template <int K, int NT>
__global__ __launch_bounds__(GEMM_THREADS)
void gemm_relu_tiled(const float* __restrict__ A, const float* __restrict__ W,
                     const float* __restrict__ bias, float* __restrict__ out,
                     int n_rows) {
  const int OW = NT * 16;
  __shared__ float sA[GEMM_ROWS * APAD];
  __shared__ float sW[KC * NT * 16];
  int tid = threadIdx.x;
  int wave = tid >> 5, lane = tid & 31;
  int half = lane >> 4, mr = lane & 15;
  int row0 = blockIdx.x * GEMM_ROWS;
  int m0 = row0 + wave * 16;

  v8f acc[NT];
#pragma unroll
  for (int j = 0; j < NT; ++j)
#pragma unroll
    for (int r = 0; r < 8; ++r) acc[j][r] = 0.0f;

  for (int k0 = 0; k0 < K; k0 += KC) {
    // stage A tile: 64 rows x 64 K (coalesced b128)
    for (int i = tid; i < GEMM_ROWS * (KC / 4); i += GEMM_THREADS) {
      int r = i >> 4;
      int c4 = (i & 15) << 2;
      int gr = row0 + r;
      gr = gr < n_rows ? gr : n_rows - 1;
      cp16_g2l(&sA[r * APAD + c4], A + (long long)gr * K + k0 + c4);
    }
    // stage W tile: 64 K x OW
    for (int i = tid; i < KC * (OW / 4); i += GEMM_THREADS) {
      int kk = i / (OW / 4);
      int c4 = (i % (OW / 4)) << 2;
      cp16_g2l(&sW[kk * OW + c4], W + (long long)(k0 + kk) * OW + c4);
    }
    stage_fence();

    const float* ar = &sA[(wave * 16 + mr) * APAD];
    for (int k = 0; k < KC; k += 4) {
      int ka = k + 2 * half;
      v2f a = *(const v2f*)(ar + ka);
#pragma unroll
      for (int j = 0; j < NT; ++j) {
        int col = j * 16 + mr;
        v2f b;
        b.x = sW[ka * OW + col];
        b.y = sW[(ka + 1) * OW + col];
        acc[j] = wmma_k4(a, b, acc[j]);
      }
    }
    __syncthreads();
  }

#pragma unroll
  for (int j = 0; j < NT; ++j)
#pragma unroll
    for (int r = 0; r < 8; ++r) {
      int row = m0 + r + 8 * half;
      int col = j * 16 + mr;
      if (row < n_rows) {
        float v = acc[j][r] + bias[col];
        out[(long long)row * OW + col] = v > 0.0f ? v : 0.0f;
      }
    }
}

// ---------------------------- edge scatter-add -----------------------------
// msg[dst] += xp[src]  (one wave per edge; L2-resident atomics)

template <int D>
__global__ void scatter_add_kernel(const float* __restrict__ xp,
                                   const int* __restrict__ src,
                                   const int* __restrict__ dst,
                                   float* __restrict__ msg, int n_edges) {
  long long t = (long long)blockIdx.x * blockDim.x + threadIdx.x;
  int e = (int)(t >> 5);
  int lane = (int)(t & 31);
  if (e >= n_edges) return;
  long long s = (long long)src[e] * D;
  long long d = (long long)dst[e] * D;
#pragma unroll
  for (int c = 0; c < D / 32; ++c) {
    int f = c * 32 + lane;
    atomicAdd(&msg[d + f], xp[s + f]);
  }
}

// ------------------------------ SAGE combine -------------------------------
// out = L2norm( (msg/max(deg,1)) @ WL + BL + x @ WR ),  out width = 64

template <int K>
__global__ __launch_bounds__(GEMM_THREADS)
void sage_combine_tiled(const float* __restrict__ X, const float* __restrict__ MSG,
                        const float* __restrict__ DEG,
                        const float* __restrict__ WL,  // K x 64
                        const float* __restrict__ BL,  // 64
                        const float* __restrict__ WR,  // K x 64
                        float* __restrict__ out,       // N x 64
                        int n_rows) {
  __shared__ float sA[GEMM_ROWS * APAD];
  __shared__ float sW[KC * HDIM];
  int tid = threadIdx.x;
  int wave = tid >> 5, lane = tid & 31;
  int half = lane >> 4, mr = lane & 15;
  int row0 = blockIdx.x * GEMM_ROWS;
  int m0 = row0 + wave * 16;
  int myrow = m0 + mr;
  myrow = myrow < n_rows ? myrow : n_rows - 1;
  float dinv = 1.0f / fmaxf(DEG[myrow], 1.0f);

  v8f acc[4];
#pragma unroll
  for (int j = 0; j < 4; ++j)
#pragma unroll
    for (int r = 0; r < 8; ++r) acc[j][r] = 0.0f;

#pragma unroll
  for (int pass = 0; pass < 2; ++pass) {
    const float* Asrc = pass ? X : MSG;
    const float* Wsrc = pass ? WR : WL;
    float scale = pass ? 1.0f : dinv;
    for (int k0 = 0; k0 < K; k0 += KC) {
      for (int i = tid; i < GEMM_ROWS * (KC / 4); i += GEMM_THREADS) {
        int r = i >> 4;
        int c4 = (i & 15) << 2;
        int gr = row0 + r;
        gr = gr < n_rows ? gr : n_rows - 1;
        cp16_g2l(&sA[r * APAD + c4], Asrc + (long long)gr * K + k0 + c4);
      }
      for (int i = tid; i < KC * (HDIM / 4); i += GEMM_THREADS) {
        int kk = i >> 4;
        int c4 = (i & 15) << 2;
        cp16_g2l(&sW[kk * HDIM + c4], Wsrc + (long long)(k0 + kk) * HDIM + c4);
      }
      stage_fence();

      const float* ar = &sA[(wave * 16 + mr) * APAD];
      for (int k = 0; k < KC; k += 4) {
        int ka = k + 2 * half;
        v2f a = *(const v2f*)(ar + ka);
        a.x *= scale;
        a.y *= scale;
#pragma unroll
        for (int j = 0; j < 4; ++j) {
          int col = j * 16 + mr;
          v2f b;
          b.x = sW[ka * HDIM + col];
          b.y = sW[(ka + 1) * HDIM + col];
          acc[j] = wmma_k4(a, b, acc[j]);
        }
      }
      __syncthreads();
    }
  }

  // bias + per-row L2 norm (rows r / r+8 live in the two 16-lane halves)
  float rs[8];
#pragma unroll
  for (int r = 0; r < 8; ++r) rs[r] = 0.0f;
#pragma unroll
  for (int j = 0; j < 4; ++j)
#pragma unroll
    for (int r = 0; r < 8; ++r) {
      float v = acc[j][r] + BL[j * 16 + mr];
      acc[j][r] = v;
      rs[r] += v * v;
    }
#pragma unroll
  for (int r = 0; r < 8; ++r) {
#pragma unroll
    for (int m = 1; m < 16; m <<= 1) rs[r] += __shfl_xor(rs[r], m, 32);
    rs[r] = 1.0f / fmaxf(sqrtf(rs[r]), 1e-12f);
  }
#pragma unroll
  for (int j = 0; j < 4; ++j)
#pragma unroll
    for (int r = 0; r < 8; ++r) {
      int row = m0 + r + 8 * half;
      if (row < n_rows)
        out[(long long)row * HDIM + j * 16 + mr] = acc[j][r] * rs[r];
    }
}

// ------------------------------ graph pooling ------------------------------

__device__ __forceinline__ int enc_ord(float v) {
  int i = __float_as_int(v);
  return i ^ ((i >> 31) & 0x7fffffff);  // order-preserving float->int
}
__device__ __forceinline__ float dec_ord(int e) {
  int i = e ^ ((e >> 31) & 0x7fffffff);
  return __int_as_float(i);
}

__global__ void pool_kernel(const float* __restrict__ x, const int* __restrict__ batch,
                            float* __restrict__ sums, int* __restrict__ maxi,
                            float* __restrict__ cnt, int n_nodes) {
  long long t = (long long)blockIdx.x * blockDim.x + threadIdx.x;
  int node = (int)(t >> 5);
  int lane = (int)(t & 31);
  if (node >= n_nodes) return;
  int g = batch[node];
#pragma unroll
  for (int c = 0; c < HDIM / 32; ++c) {
    int f = c * 32 + lane;
    float v = x[(long long)node * HDIM + f];
    atomicAdd(&sums[g * HDIM + f], v);
    atomicMax(&maxi[g * HDIM + f], enc_ord(v));
  }
  if (lane == 0) atomicAdd(&cnt[g], 1.0f);
}

__global__ void final_kernel(const float* __restrict__ sums, const int* __restrict__ maxi,
                             const float* __restrict__ cnt, const float* __restrict__ pw,
                             const float* __restrict__ pb, float* __restrict__ out) {
  __shared__ float ssq[HDIM];
  __shared__ float sdot[HDIM];
  int g = blockIdx.x;
  int f = threadIdx.x;
  float v = dec_ord(maxi[g * HDIM + f]) + sums[g * HDIM + f] / cnt[g];
  ssq[f] = v * v;
  sdot[f] = v * pw[f];
  __syncthreads();
  if (f == 0) {
    float s = 0.0f, dt = 0.0f;
    for (int t = 0; t < HDIM; ++t) { s += ssq[t]; dt += sdot[t]; }
    out[g] = dt / fmaxf(sqrtf(s), 1e-30f) + pb[0];
  }
}

// ------------------------------- launcher ----------------------------------

extern "C" void kernel_launch(void* const* d_in, const int* in_sizes, int n_in,
                              void* d_out, int out_size, void* d_ws, size_t ws_size,
                              hipStream_t stream) {
  const float* node_feat = (const float*)d_in[0];
  const float* cfg       = (const float*)d_in[1];
  const int*   opcode    = (const int*)d_in[2];
  const int*   edge      = (const int*)d_in[3];
  const int*   batch     = (const int*)d_in[4];
  const float* op_emb    = (const float*)d_in[5];
  const float* lin_w     = (const float*)d_in[6];
  const float* lin_b     = (const float*)d_in[7];
  const float* post_w    = (const float*)d_in[8];
  const float* post_b    = (const float*)d_in[9];
  const float* pw[3] = {(const float*)d_in[10], (const float*)d_in[15], (const float*)d_in[20]};
  const float* pb[3] = {(const float*)d_in[11], (const float*)d_in[16], (const float*)d_in[21]};
  const float* wl[3] = {(const float*)d_in[12], (const float*)d_in[17], (const float*)d_in[22]};
  const float* bl[3] = {(const float*)d_in[13], (const float*)d_in[18], (const float*)d_in[23]};
  const float* wr[3] = {(const float*)d_in[14], (const float*)d_in[19], (const float*)d_in[24]};

  int n_nodes = in_sizes[2];
  int n_edges = in_sizes[3] / 2;
  const int* src = edge;
  const int* dst = edge + n_edges;

  // ---- workspace layout (256B aligned) ----
  char* w = (char*)d_ws;
  size_t SZ128 = (((size_t)n_nodes * LOUT * sizeof(float)) + 255) & ~(size_t)255;
  size_t SZ192 = (((size_t)n_nodes * KEMB * sizeof(float)) + 255) & ~(size_t)255;
  float* P0 = (float*)(w);                 // x0 (N x 128)
  float* P1 = (float*)(w + SZ128);         // xp0 (N x 128)
  float* XC = (float*)(w + 2 * SZ128);     // concat features (N x 192)
  float* M  = XC;                          // msg buffer aliases XC (XC dead after embed)
  size_t degsz = (((size_t)n_nodes * sizeof(float)) + 255) & ~(size_t)255;
  float* deg  = (float*)(w + 2 * SZ128 + SZ192);
  float* sums = (float*)((char*)deg + degsz);
  int*   maxi = (int*)((char*)sums + NGRAPH * HDIM * sizeof(float));
  float* cnt  = (float*)((char*)maxi + NGRAPH * HDIM * sizeof(int));
  float* W2   = (float*)((char*)cnt + ((NGRAPH * sizeof(float) + 255) & ~(size_t)255));
  // 64-wide activations reuse the freed 128-wide regions
  float* Q0 = P1;                              // x1
  float* Q1 = P1 + (size_t)n_nodes * HDIM;     // xp1
  float* R0 = P0;                              // x2
  float* R1 = P0 + (size_t)n_nodes * HDIM;     // xp2

  int gemm_blocks = (n_nodes + GEMM_ROWS - 1) / GEMM_ROWS;
  int edge_blocks = (int)(((long long)n_edges * 32 + 255) / 256);
  int node_blocks = (int)(((long long)n_nodes * 32 + 255) / 256);

  // degree (layer-invariant within one call)
  fill_f32_kernel<<<1024, 256, 0, stream>>>(deg, 0.0f, n_nodes);
  degree_kernel<<<(n_edges + 255) / 256, 256, 0, stream>>>(dst, deg, n_edges);

  // input prep: padded concat features + padded embed weights
  concat_pad_kernel<<<n_nodes, KEMB, 0, stream>>>(node_feat, cfg, opcode, op_emb, XC);
  pad_w_kernel<<<(KEMB * LOUT + 255) / 256, 256, 0, stream>>>(lin_w, W2);

  // x0 = relu(XC @ W2 + lin_b)
  gemm_relu_tiled<KEMB, 8><<<gemm_blocks, GEMM_THREADS, 0, stream>>>(
      XC, W2, lin_b, P0, n_nodes);

  // ---- layer 0 (d=128 -> 64) ----
  gemm_relu_tiled<128, 8><<<gemm_blocks, GEMM_THREADS, 0, stream>>>(
      P0, pw[0], pb[0], P1, n_nodes);
  fill_f32_kernel<<<2048, 256, 0, stream>>>(M, 0.0f, (long long)n_nodes * 128);
  scatter_add_kernel<128><<<edge_blocks, 256, 0, stream>>>(P1, src, dst, M, n_edges);
  sage_combine_tiled<128><<<gemm_blocks, GEMM_THREADS, 0, stream>>>(
      P0, M, deg, wl[0], bl[0], wr[0], Q0, n_nodes);

  // ---- layer 1 (d=64 -> 64) ----
  gemm_relu_tiled<64, 4><<<gemm_blocks, GEMM_THREADS, 0, stream>>>(
      Q0, pw[1], pb[1], Q1, n_nodes);
  fill_f32_kernel<<<2048, 256, 0, stream>>>(M, 0.0f, (long long)n_nodes * 64);
  scatter_add_kernel<64><<<edge_blocks, 256, 0, stream>>>(Q1, src, dst, M, n_edges);
  sage_combine_tiled<64><<<gemm_blocks, GEMM_THREADS, 0, stream>>>(
      Q0, M, deg, wl[1], bl[1], wr[1], R0, n_nodes);

  // ---- layer 2 (d=64 -> 64) ----
  gemm_relu_tiled<64, 4><<<gemm_blocks, GEMM_THREADS, 0, stream>>>(
      R0, pw[2], pb[2], R1, n_nodes);
  fill_f32_kernel<<<2048, 256, 0, stream>>>(M, 0.0f, (long long)n_nodes * 64);
  scatter_add_kernel<64><<<edge_blocks, 256, 0, stream>>>(R1, src, dst, M, n_edges);
  sage_combine_tiled<64><<<gemm_blocks, GEMM_THREADS, 0, stream>>>(
      R0, M, deg, wl[2], bl[2], wr[2], Q0, n_nodes);

  // ---- pooling + head ----
  pool_init_kernel<<<(NGRAPH * HDIM + 255) / 256, 256, 0, stream>>>(sums, maxi, cnt);
  pool_kernel<<<node_blocks, 256, 0, stream>>>(Q0, batch, sums, maxi, cnt, n_nodes);
  final_kernel<<<NGRAPH, HDIM, 0, stream>>>(sums, maxi, cnt, post_w, post_b, (float*)d_out);

  (void)n_in; (void)out_size; (void)ws_size;
}